// StructuredLightProjectionLayer_833223655542
// MI455X (gfx1250) — compile-verified
//
#include <hip/hip_runtime.h>
#include <hip/hip_bf16.h>
#include <stdint.h>
#include <math.h>

#define BB  16
#define HH  720
#define WW  1280
#define PHH 720
#define PWW 1344
#define PAT_OFF 48.0f
#define NEG_BIG -3.0e38f

typedef float v2f __attribute__((ext_vector_type(2)));
typedef float v8f __attribute__((ext_vector_type(8)));

// Deterministic per-pixel Gaussian read noise: hash(y,x) -> Box-Muller.
// (JAX threefry with key 42 is not reproducible in HIP; this is a stand-in
//  with matching mean/stddev, deterministic across calls.)
__device__ __forceinline__ float read_noise(int y, int x) {
  unsigned h = (unsigned)(y * WW + x) * 2654435761u + 0x9E3779B9u;
  h ^= h >> 16; h *= 0x85EBCA6Bu; h ^= h >> 13; h *= 0xC2B2AE35u; h ^= h >> 16;
  unsigned h2 = h * 0x27d4eb2du + 0x165667b1u;
  h2 ^= h2 >> 15; h2 *= 0x2c1b3c6du; h2 ^= h2 >> 12;
  float u1 = ((h  & 0xFFFFFFu) + 1u) * (1.0f / 16777217.0f);   // (0,1]
  float u2 = (h2 & 0xFFFFFFu) * (1.0f / 16777216.0f);          // [0,1)
  float r = sqrtf(-2.0f * logf(u1));
  float n = r * cosf(6.28318530718f * u2);
  return 30.0f / 1024.0f + (10.0f / 1024.0f) * n;
}

// One 256-thread block per (b,y) row. PASS==1: accumulate per-row sum of
// un-normalized recv. PASS==2: write composed output using per-batch scale.
template <int PASS>
__global__ __launch_bounds__(256) void slp_row_kernel(
    const float* __restrict__ calib, const float* __restrict__ depth,
    const float* __restrict__ ambient, const float* __restrict__ albedo,
    const float* __restrict__ pattern, float* __restrict__ row_sums,
    const float* __restrict__ scale, float* __restrict__ composed)
{
  __shared__ __align__(16) float s_depth[WW];  // staged via async global->LDS
  __shared__ float s_suf[WW];                  // inclusive suffix max of d[x]-x
  __shared__ float s_cm[257];
  __shared__ float s_red[8];

  const int row = blockIdx.x;          // b*HH + y
  const int b = row / HH;
  const int y = row - b * HH;
  const int t = threadIdx.x;

  // Per-batch calibration (uniform across block -> scalar loads).
  const float fx = calib[b * 6 + 0];
  const float fy = calib[b * 6 + 1];
  const float cx = calib[b * 6 + 2];
  const float cy = calib[b * 6 + 3];
  const float bl = calib[b * 6 + 4];
  const float zr = calib[b * 6 + 5];
  const float fxb  = fx * bl;
  const float dref = fxb / zr;
  const float rfx  = 1.0f / fx;
  const float rfy  = 1.0f / fy;

  const float* __restrict__ drow = depth + (size_t)row * WW;

  // ---- CDNA5 async copy: depth row (1280 f32 = 320 b128) into LDS ----
  {
    unsigned lbase = (unsigned)(uintptr_t)&s_depth[0];  // low 32 bits = LDS offset
    const float* gp = drow + 4 * t;
    unsigned lo = lbase + 16u * t;
    asm volatile("global_load_async_to_lds_b128 %0, %1, off"
                 :: "v"(lo), "v"(gp) : "memory");
    if (t < (WW / 4 - 256)) {                            // remaining 64 vec4s
      const float* gp2 = drow + 4 * (256 + t);
      unsigned lo2 = lbase + 16u * (256 + t);
      asm volatile("global_load_async_to_lds_b128 %0, %1, off"
                   :: "v"(lo2), "v"(gp2) : "memory");
    }
  }
  asm volatile("s_wait_asynccnt 0x0" ::: "memory");
  __syncthreads();

  // ---- Suffix-max scan of g[x] = disparity[x] - x (shadow test) ----
  // Thread t owns contiguous chunk [5t, 5t+5): local suffix max in registers.
  float lsuf[5];
  {
    float m = NEG_BIG;
    #pragma unroll
    for (int k = 4; k >= 0; --k) {
      int x = t * 5 + k;
      float z = s_depth[x];
      float g = fxb / z - dref - (float)x;
      m = fmaxf(m, g);
      lsuf[k] = m;
    }
    s_cm[t] = m;
  }
  __syncthreads();
  // Hillis-Steele inclusive suffix-max over the 256 chunk maxima.
  for (int off = 1; off < 256; off <<= 1) {
    float v = s_cm[t];
    float w = (t + off < 256) ? s_cm[t + off] : NEG_BIG;
    __syncthreads();
    s_cm[t] = fmaxf(v, w);
    __syncthreads();
  }
  {
    float tail = (t < 255) ? s_cm[t + 1] : NEG_BIG;  // suffix max of later chunks
    #pragma unroll
    for (int k = 0; k < 5; ++k) s_suf[t * 5 + k] = fmaxf(lsuf[k], tail);
  }
  __syncthreads();

  // ---- Per-pixel pipeline (coalesced layout x = t + 256k) ----
  const float* __restrict__ alrow = albedo  + (size_t)row * WW;
  const float* __restrict__ amrow = ambient + (size_t)row * WW;
  const float* __restrict__ prow  = pattern + (size_t)y * PWW;
  float* __restrict__ orow = composed + (size_t)row * WW;
  float sc = (PASS == 2) ? scale[b] : 0.0f;
  float acc = 0.0f;
  #pragma unroll
  for (int k = 0; k < 5; ++k) {
    int x = t + 256 * k;
    float z = s_depth[x];
    float d = fxb / z - dref;                       // disparity
    float g = d - (float)x;
    float mask = fminf(fmaxf(g - s_suf[x] + 1.0f, 0.0f), 1.0f);
    // 1-D bilinear sample of the pattern row
    float px = (float)x - d + PAT_OFF;
    float xf = floorf(px);
    float fr = px - xf;
    int x0 = (int)xf;
    float pf = prow[x0];
    float pc = prow[x0 + 1];
    float proj = pf + (pc - pf) * fr;
    // inverse-square falloff
    float p3x = ((float)x - cx) * z * rfx;
    float p3y = ((float)y - cy) * z * rfy;
    float den = p3x * p3x + p3y * p3y + z * z;
    float fall = fminf(1.0f / den, 2.0f);
    float r = proj * mask * alrow[x] * fall;
    if (PASS == 1) {
      acc += r;
    } else {
      orow[x] = amrow[x] + r * sc + read_noise(y, x);
    }
  }

  if (PASS == 1) {
    // wave32 butterfly + cross-wave LDS reduce -> per-row sum
    #pragma unroll
    for (int o = 16; o > 0; o >>= 1) acc += __shfl_xor(acc, o, 32);
    if ((t & 31) == 0) s_red[t >> 5] = acc;
    __syncthreads();
    if (t == 0) {
      float s = 0.0f;
      #pragma unroll
      for (int i = 0; i < 8; ++i) s += s_red[i];
      row_sums[row] = s;
    }
  }
}

// One wave per batch: reduce 720 row sums with V_WMMA_F32_16X16X4_F32.
// D = A x ones(4x16) + C sums 64 f32 per issue; D is replicated across the
// 16 columns, so a full-lane butterfly over-counts by exactly 16x.
__global__ __launch_bounds__(32) void slp_mean_wmma_kernel(
    const float* __restrict__ row_sums, float* __restrict__ scale)
{
  const int b = blockIdx.x;
  const int lane = threadIdx.x;  // 0..31, all active (WMMA needs EXEC all-1s)
  const float* __restrict__ src = row_sums + b * HH;

  v8f c = {0.f, 0.f, 0.f, 0.f, 0.f, 0.f, 0.f, 0.f};
  v2f onesv; onesv[0] = 1.0f; onesv[1] = 1.0f;
  #pragma unroll
  for (int ch = 0; ch < 12; ++ch) {              // 12*64 = 768 >= 720
    int i0 = ch * 64 + lane * 2;
    v2f a;
    a[0] = (i0     < HH) ? src[i0]     : 0.0f;   // select, no divergence
    a[1] = (i0 + 1 < HH) ? src[i0 + 1] : 0.0f;
    c = __builtin_amdgcn_wmma_f32_16x16x4_f32(
        /*neg_a=*/false, a, /*neg_b=*/false, onesv,
        /*c_mod=*/(short)0, c, /*reuse_a=*/false, /*reuse_b=*/false);
  }
  float s = c[0] + c[1] + c[2] + c[3] + c[4] + c[5] + c[6] + c[7];
  #pragma unroll
  for (int o = 16; o > 0; o >>= 1) s += __shfl_xor(s, o, 32);
  s *= (1.0f / 16.0f);                           // undo 16-column replication
  if (lane == 0) scale[b] = 0.35f * (float)(HH * WW) / s;
}

extern "C" void kernel_launch(void* const* d_in, const int* in_sizes, int n_in,
                              void* d_out, int out_size, void* d_ws, size_t ws_size,
                              hipStream_t stream) {
  const float* calib   = (const float*)d_in[0];
  const float* depth   = (const float*)d_in[1];
  // d_in[2] ("unused") is never read by the reference.
  const float* ambient = (const float*)d_in[3];
  const float* albedo  = (const float*)d_in[4];
  const float* pattern = (const float*)d_in[5];
  float* out = (float*)d_out;

  float* row_sums = (float*)d_ws;          // BB*HH floats
  float* scale    = row_sums + BB * HH;    // BB floats

  // Output 0: the pattern itself, shape (1, PH, PW).
  hipMemcpyAsync(out, pattern, (size_t)PHH * PWW * sizeof(float),
                 hipMemcpyDeviceToDevice, stream);

  float* composed = out + (size_t)PHH * PWW;
  dim3 grid(BB * HH), block(256);
  slp_row_kernel<1><<<grid, block, 0, stream>>>(
      calib, depth, ambient, albedo, pattern, row_sums, scale, composed);
  slp_mean_wmma_kernel<<<dim3(BB), dim3(32), 0, stream>>>(row_sums, scale);
  slp_row_kernel<2><<<grid, block, 0, stream>>>(
      calib, depth, ambient, albedo, pattern, row_sums, scale, composed);
}